// DomainAdaptiveBatchNorm_57415122813330
// MI455X (gfx1250) — compile-verified
//
#include <hip/hip_runtime.h>

// ---------------------------------------------------------------------------
// DomainAdaptiveBatchNorm on SPD(32) matrices, fp32, for gfx1250 (MI455X).
// One wave32 per matrix; parallel Jacobi eigensolver in LDS (stride-33 pad);
// all 32x32x32 matmuls via V_WMMA_F32_16X16X4_F32.
// ---------------------------------------------------------------------------

#define LD     33        // padded LDS leading dimension (conflict-free)
#define NMAT   32768     // N*q
#define SWEEPS 10        // Jacobi sweeps (quadratic convergence, 32x32 SPD)
#define EPSV   1e-5f

// workspace layout (floats)
#define WS_G0   0        // arithmetic mean accumulator
#define WS_GS   1024     // G^{1/2}
#define WS_GIS  2048     // G^{-1/2}
#define WS_GT   3072     // tangent-space mean accumulator
#define WS_GH   4096     // final mean^{-1/2}
#define WS_BH   5120     // B^{1/2}
#define WS_VAR  6144
#define WS_P    6145
#define WS_END  6656

typedef float v2f __attribute__((ext_vector_type(2)));
typedef float v8f __attribute__((ext_vector_type(8)));

// ---------------- global <-> LDS 32x32 (row-major, stride LD) ---------------
__device__ __forceinline__ void ld_mat(const float* g, float* s, int lane) {
  for (int i = lane; i < 1024; i += 32) s[(i >> 5) * LD + (i & 31)] = g[i];
  __syncthreads();
}
__device__ __forceinline__ void st_mat(const float* s, float* g, int lane) {
  __syncthreads();
  for (int i = lane; i < 1024; i += 32) g[i] = s[(i >> 5) * LD + (i & 31)];
}

// ---------------- C = A * B (or A * B^T), 32x32 fp32, one wave --------------
// Uses V_WMMA_F32_16X16X4_F32: A 16x4 (v2f), B 4x16 (v2f), C/D 16x16 (v8f).
// A layout: lanes 0-15 M=0..15 {K=k0,k0+1}; lanes 16-31 {K=k0+2,k0+3}.
// B layout: VGPR v: lanes 0-15 row K=v, lanes 16-31 row K=v+2.
// C layout: VGPR v: lanes 0-15 M=v, lanes 16-31 M=v+8; N = lane&15.
__device__ __forceinline__ void mm32(const float* A, const float* B, float* C,
                                     bool transB, int lane) {
  __syncthreads();
  const int half = lane >> 4;
  const int lm   = lane & 15;
  v8f acc[2][2];
#pragma unroll
  for (int i = 0; i < 2; ++i)
#pragma unroll
    for (int j = 0; j < 2; ++j) acc[i][j] = (v8f)0.0f;

#pragma unroll
  for (int kk = 0; kk < 8; ++kk) {
    const int k0 = kk * 4 + half * 2;
    v2f af[2], bf[2];
#pragma unroll
    for (int i = 0; i < 2; ++i) {
      const float* ap = A + (i * 16 + lm) * LD + k0;
      af[i].x = ap[0];
      af[i].y = ap[1];
    }
#pragma unroll
    for (int j = 0; j < 2; ++j) {
      if (!transB) {
        bf[j].x = B[(k0    ) * LD + j * 16 + lm];
        bf[j].y = B[(k0 + 1) * LD + j * 16 + lm];
      } else {
        const float* bp = B + (j * 16 + lm) * LD + k0;
        bf[j].x = bp[0];
        bf[j].y = bp[1];
      }
    }
#pragma unroll
    for (int i = 0; i < 2; ++i)
#pragma unroll
      for (int j = 0; j < 2; ++j)
        acc[i][j] = __builtin_amdgcn_wmma_f32_16x16x4_f32(
            false, af[i], false, bf[j], (short)0, acc[i][j], false, false);
  }
  __syncthreads();
#pragma unroll
  for (int i = 0; i < 2; ++i)
#pragma unroll
    for (int j = 0; j < 2; ++j)
#pragma unroll
      for (int v = 0; v < 8; ++v) {
        const int row = i * 16 + v + half * 8;
        C[row * LD + j * 16 + lm] = acc[i][j][v];
      }
  __syncthreads();
}

// -------- parallel cyclic Jacobi eigensolver, 32x32 symmetric, one wave -----
// A destroyed (diag -> eigenvalues), V <- eigenvectors (columns).
// Lane c owns column c. Round-robin tournament: 16 disjoint pairs per round.
__device__ __forceinline__ int pos2player(int pos, int rnd) {
  return (pos == 0) ? 0 : 1 + ((pos - 1 + rnd) % 31);
}
__device__ __forceinline__ void jacobi32(float* A, float* V, float* csC,
                                         float* csS, int lane) {
  for (int r = 0; r < 32; ++r) V[r * LD + lane] = (r == lane) ? 1.0f : 0.0f;
  __syncthreads();
  for (int sweep = 0; sweep < SWEEPS; ++sweep) {
    for (int rnd = 0; rnd < 31; ++rnd) {
      // my position & partner in this round
      const int myPos = (lane == 0) ? 0 : 1 + ((lane - 1 - rnd + 31 * 32) % 31);
      const int partner = pos2player(31 - myPos, rnd);
      const int p = (lane < partner) ? lane : partner;
      const int q = (lane < partner) ? partner : lane;
      const float app = A[p * LD + p];
      const float aqq = A[q * LD + q];
      const float apq = A[p * LD + q];
      float cc = 1.0f, ss = 0.0f;
      if (fabsf(apq) > 1e-12f) {
        const float tau = (aqq - app) / (2.0f * apq);
        const float t = copysignf(1.0f, tau) / (fabsf(tau) + sqrtf(1.0f + tau * tau));
        cc = rsqrtf(1.0f + t * t);
        ss = t * cc;
      }
      csC[lane] = cc;
      csS[lane] = ss;
      __syncthreads();
      // column phase: A <- A*J, V <- V*J (lane writes only its own column;
      // same-wave LDS ops are in-order, loads of a row precede its stores)
      const bool isP = (lane == p);
#pragma unroll 4
      for (int r = 0; r < 32; ++r) {
        const float ap_ = A[r * LD + p], aq_ = A[r * LD + q];
        A[r * LD + lane] = isP ? (cc * ap_ - ss * aq_) : (ss * ap_ + cc * aq_);
        const float vp_ = V[r * LD + p], vq_ = V[r * LD + q];
        V[r * LD + lane] = isP ? (cc * vp_ - ss * vq_) : (ss * vp_ + cc * vq_);
      }
      __syncthreads();
      // row phase: A <- J^T*A (each lane touches only its own column)
#pragma unroll 4
      for (int k = 0; k < 16; ++k) {
        const int u  = pos2player(k, rnd);
        const int v2 = pos2player(31 - k, rnd);
        const int pp = (u < v2) ? u : v2;
        const int qq = (u < v2) ? v2 : u;
        const float ck = csC[pp], sk = csS[pp];
        const float x = A[pp * LD + lane], y = A[qq * LD + lane];
        A[pp * LD + lane] = ck * x - sk * y;
        A[qq * LD + lane] = sk * x + ck * y;
      }
      __syncthreads();
    }
  }
}

// OUT = V * diag(f_lane) * V^T     (T = scratch buffer, distinct from V/OUT)
__device__ __forceinline__ void recon(const float* V, float f, float* T,
                                      float* OUT, int lane) {
  __syncthreads();
  for (int r = 0; r < 32; ++r) T[r * LD + lane] = V[r * LD + lane] * f;
  mm32(T, V, OUT, /*transB=*/true, lane);
}

// ------------------------------- kernels ------------------------------------

__global__ void k_zero(float* ws) {
  const int i = blockIdx.x * blockDim.x + threadIdx.x;
  if (i < WS_END) ws[i] = 0.0f;
}

// arithmetic mean of 32768 matrices -> ws[WS_G0]
__global__ void k_mean(const float* __restrict__ X, float* ws) {
  const int t = threadIdx.x;          // 256 threads, 4 elems each
  const int e = t * 4;
  const size_t m0 = (size_t)blockIdx.x * 128;  // 256 blocks x 128 matrices
  float a0 = 0.f, a1 = 0.f, a2 = 0.f, a3 = 0.f;
  for (int m = 0; m < 128; ++m) {
    const float* Xm = X + (m0 + m) * 1024 + e;
    a0 += Xm[0]; a1 += Xm[1]; a2 += Xm[2]; a3 += Xm[3];
  }
  const float sc = 1.0f / (float)NMAT;
  atomicAdd(&ws[WS_G0 + e + 0], a0 * sc);
  atomicAdd(&ws[WS_G0 + e + 1], a1 * sc);
  atomicAdd(&ws[WS_G0 + e + 2], a2 * sc);
  atomicAdd(&ws[WS_G0 + e + 3], a3 * sc);
}

// eig(G0) -> Gs = G^{1/2}, Gis = G^{-1/2}
__global__ __launch_bounds__(32) void k_prep(float* ws) {
  __shared__ float A[32 * LD], V[32 * LD], T1[32 * LD], T2[32 * LD];
  __shared__ float csC[32], csS[32];
  const int lane = threadIdx.x;
  ld_mat(ws + WS_G0, A, lane);
  jacobi32(A, V, csC, csS, lane);
  const float w = fmaxf(A[lane * LD + lane], 1e-20f);
  recon(V, sqrtf(w), T1, T2, lane);
  st_mat(T2, ws + WS_GS, lane);
  recon(V, rsqrtf(w), T1, T2, lane);
  st_mat(T2, ws + WS_GIS, lane);
}

// per matrix: L = log(Gis * X * Gis), accumulate mean into ws[WS_GT]
__global__ __launch_bounds__(32) void k_karcher(const float* __restrict__ X,
                                                float* ws) {
  __shared__ float A[32 * LD], V[32 * LD], T1[32 * LD], T2[32 * LD];
  __shared__ float csC[32], csS[32];
  const int lane = threadIdx.x;
  const size_t m = blockIdx.x;
  ld_mat(X + m * 1024, T1, lane);       // T1 = X
  ld_mat(ws + WS_GIS, T2, lane);        // T2 = Gis
  mm32(T2, T1, A, false, lane);         // A  = Gis*X
  mm32(A, T2, T1, false, lane);         // T1 = Gis*X*Gis
  jacobi32(T1, V, csC, csS, lane);
  const float w = fmaxf(T1[lane * LD + lane], 1e-20f);
  recon(V, logf(w), A, T2, lane);       // T2 = log(Y)
  const float sc = 1.0f / (float)NMAT;
  for (int r = 0; r < 32; ++r)
    atomicAdd(&ws[WS_GT + r * 32 + lane], T2[r * LD + lane] * sc);
}

// G1 = Gs*exp(GT)*Gs ; Gh = G1^{-1/2} ; Bh = B^{1/2}
__global__ __launch_bounds__(32) void k_finalize(float* ws,
                                                 const float* __restrict__ B) {
  __shared__ float A[32 * LD], V[32 * LD], T1[32 * LD], T2[32 * LD];
  __shared__ float csC[32], csS[32];
  const int lane = threadIdx.x;
  ld_mat(ws + WS_GT, A, lane);
  jacobi32(A, V, csC, csS, lane);
  const float wt = A[lane * LD + lane];
  recon(V, expf(wt), T1, T2, lane);     // T2 = exp(GT)
  ld_mat(ws + WS_GS, T1, lane);         // T1 = Gs
  mm32(T1, T2, A, false, lane);         // A  = Gs*exp(GT)
  mm32(A, T1, T2, false, lane);         // T2 = G1
  jacobi32(T2, V, csC, csS, lane);
  const float w1 = fmaxf(T2[lane * LD + lane], 1e-20f);
  recon(V, rsqrtf(w1), T1, A, lane);    // A = Gh
  st_mat(A, ws + WS_GH, lane);
  ld_mat(B, T1, lane);
  jacobi32(T1, V, csC, csS, lane);
  const float wb = fmaxf(T1[lane * LD + lane], 1e-20f);
  recon(V, sqrtf(wb), T2, A, lane);     // A = Bh
  st_mat(A, ws + WS_BH, lane);
}

// per matrix: var contribution = ||log(Gh*X*Gh)||_F^2 = sum log(lambda)^2
__global__ __launch_bounds__(32) void k_var(const float* __restrict__ X,
                                            float* ws) {
  __shared__ float A[32 * LD], V[32 * LD], T1[32 * LD], T2[32 * LD];
  __shared__ float csC[32], csS[32];
  const int lane = threadIdx.x;
  const size_t m = blockIdx.x;
  ld_mat(X + m * 1024, T1, lane);
  ld_mat(ws + WS_GH, T2, lane);
  mm32(T2, T1, A, false, lane);
  mm32(A, T2, T1, false, lane);         // T1 = Xw
  jacobi32(T1, V, csC, csS, lane);
  const float w = fmaxf(T1[lane * LD + lane], 1e-20f);
  const float l = logf(w);
  csC[lane] = l * l;
  __syncthreads();
  if (lane == 0) {
    float t = 0.f;
    for (int i = 0; i < 32; ++i) t += csC[i];
    atomicAdd(&ws[WS_VAR], t * (1.0f / (float)NMAT));
  }
}

__global__ void k_p(float* ws) {
  ws[WS_P] = sqrtf(1.0f / (ws[WS_VAR] + EPSV));
}

// per matrix: Out = Bh * ( R * (Gh*X*Gh)^p * R^T ) * Bh
__global__ __launch_bounds__(32) void k_out(const float* __restrict__ X,
                                            const float* __restrict__ R,
                                            const float* __restrict__ ws,
                                            float* __restrict__ out) {
  __shared__ float A[32 * LD], V[32 * LD], T1[32 * LD], T2[32 * LD];
  __shared__ float csC[32], csS[32];
  const int lane = threadIdx.x;
  const size_t m = blockIdx.x;
  ld_mat(X + m * 1024, T1, lane);
  ld_mat(ws + WS_GH, T2, lane);
  mm32(T2, T1, A, false, lane);
  mm32(A, T2, T1, false, lane);         // T1 = Xw
  jacobi32(T1, V, csC, csS, lane);
  const float w = fmaxf(T1[lane * LD + lane], 1e-20f);
  const float p = ws[WS_P];
  const float fw = expf(p * logf(w));   // w^p
  recon(V, fw, A, T2, lane);            // T2 = Xs
  ld_mat(R, T1, lane);
  mm32(T1, T2, A, false, lane);         // A  = R*Xs
  mm32(A, T1, T2, true, lane);          // T2 = R*Xs*R^T
  ld_mat(ws + WS_BH, T1, lane);
  mm32(T1, T2, A, false, lane);         // A  = Bh*Xr
  mm32(A, T1, T2, false, lane);         // T2 = Bh*Xr*Bh
  st_mat(T2, out + m * 1024, lane);
}

// ---------------------------------------------------------------------------
extern "C" void kernel_launch(void* const* d_in, const int* in_sizes, int n_in,
                              void* d_out, int out_size, void* d_ws,
                              size_t ws_size, hipStream_t stream) {
  const float* X = (const float*)d_in[0];   // (1024,32,32,32) fp32
  const float* R = (const float*)d_in[1];   // (32,32)
  const float* B = (const float*)d_in[2];   // (32,32)
  float* out = (float*)d_out;
  float* ws  = (float*)d_ws;

  k_zero<<<(WS_END + 255) / 256, 256, 0, stream>>>(ws);
  k_mean<<<256, 256, 0, stream>>>(X, ws);
  k_prep<<<1, 32, 0, stream>>>(ws);
  k_karcher<<<NMAT, 32, 0, stream>>>(X, ws);
  k_finalize<<<1, 32, 0, stream>>>(ws, B);
  k_var<<<NMAT, 32, 0, stream>>>(X, ws);
  k_p<<<1, 1, 0, stream>>>(ws);
  k_out<<<NMAT, 32, 0, stream>>>(X, R, ws, out);
}